// VanDerWallsSurface_20255065768638
// MI455X (gfx1250) — compile-verified
//
#include <hip/hip_runtime.h>
#include <stdint.h>

// Van der Waals surface voxelization for MI455X (gfx1250, wave32).
// Phase 1: init accumulators (-inf/+inf/0 + zero counters).
// Phase 2: scatter — atom data staged per block via async global->LDS DMA
//          (global_load_async_to_lds_b128, ASYNCcnt), per-atom derived values
//          precomputed once into LDS, then one (atom,offset) pair per thread
//          with native L2 f32 atomics (global_atomic_{max_num,min_num,add}_f32).
// Phase 3: finalize — mask empty voxels, mean = sum/cnt.

#define VOL 128
#define K_OFF 125            // 5x5x5 neighborhood
#define ATOMS_PER_BLOCK 32
#define SCATTER_THREADS 256

// ------------------------------------------------------------- scatter ------
__global__ void __launch_bounds__(SCATTER_THREADS)
vdw_scatter_kernel(const float* __restrict__ cr,   // (B,N,4) coords+radius
                   const float* __restrict__ ft,   // (B,N,3) features
                   float* __restrict__ out,        // (B,V,V,V,3) accumulators
                   unsigned* __restrict__ cnt,     // (B*V^3) counters
                   int NperB, int BN) {
    // raw async staging target: cr chunk (128 dwords) + ft chunk (96 dwords)
    __shared__ __align__(16) float s_raw[ATOMS_PER_BLOCK * 4 + ATOMS_PER_BLOCK * 3];
    // derived per-atom data: cx,cy,cz,r2,f0,f1,f2, pack(bx|by<<8|bz<<16|b<<24)
    __shared__ __align__(16) float s_atom[ATOMS_PER_BLOCK * 8];

    const int tid = threadIdx.x;
    const int a0  = blockIdx.x * ATOMS_PER_BLOCK;
    int atoms = BN - a0;
    if (atoms > ATOMS_PER_BLOCK) atoms = ATOMS_PER_BLOCK;

    const int n_cr = ATOMS_PER_BLOCK * 4;  // 128 dwords staged for cr

    // Opaque never-taken store: prevents the compiler from treating s_raw as
    // never-written (the real writes happen via the async DMA below).
    if (__builtin_expect(BN == -1, 0)) s_raw[tid % (n_cr + ATOMS_PER_BLOCK * 3)] = 0.0f;

    // --- stage this block's atom data into LDS: 56 lanes x async b128 -------
    // chunks 0..31  : cr  (16B each, atom c)        -> s_raw[0   + 4c]
    // chunks 32..55 : ft  (16B each, dwords 4j..)   -> s_raw[128 + 4j]
    if (tid < 56) {
        uint32_t lds_off;
        uint64_t gaddr;
        bool issue;
        if (tid < 32) {
            lds_off = (uint32_t)(uintptr_t)(&s_raw[tid * 4]);
            gaddr   = (uint64_t)(uintptr_t)(cr + (size_t)a0 * 4 + tid * 4);
            issue   = (tid < atoms);
        } else {
            const int j = tid - 32;                 // 0..23
            lds_off = (uint32_t)(uintptr_t)(&s_raw[n_cr + j * 4]);
            gaddr   = (uint64_t)(uintptr_t)(ft + (size_t)a0 * 3 + j * 4);
            issue   = (j * 4 < atoms * 3);
        }
        if (issue) {
            asm volatile("global_load_async_to_lds_b128 %0, %1, off"
                         :: "v"(lds_off), "v"(gaddr)
                         : "memory");
        }
    }
    asm volatile("s_wait_asynccnt 0x0" ::: "memory");
    __syncthreads();

    // --- per-atom derived values (once per atom instead of once per pair) ---
    if (tid < atoms) {
        const float cx = s_raw[tid * 4 + 0];
        const float cy = s_raw[tid * 4 + 1];
        const float cz = s_raw[tid * 4 + 2];
        float       r  = s_raw[tid * 4 + 3];
        // radius quantization: round(r*1000)/1000, RNE, no contraction
        r = __fdiv_rn(rintf(__fmul_rn(r, 1000.0f)), 1000.0f);
        const float r2 = __fmul_rn(r, r);
        // base voxel = round-half-even(coord); coords in [4,124] -> fits u8
        const unsigned bx = (unsigned)(int)rintf(cx);
        const unsigned by = (unsigned)(int)rintf(cy);
        const unsigned bz = (unsigned)(int)rintf(cz);
        const unsigned b  = (unsigned)((a0 + tid) / NperB);
        s_atom[tid * 8 + 0] = cx;
        s_atom[tid * 8 + 1] = cy;
        s_atom[tid * 8 + 2] = cz;
        s_atom[tid * 8 + 3] = r2;
        s_atom[tid * 8 + 4] = s_raw[n_cr + tid * 3 + 0];
        s_atom[tid * 8 + 5] = s_raw[n_cr + tid * 3 + 1];
        s_atom[tid * 8 + 6] = s_raw[n_cr + tid * 3 + 2];
        s_atom[tid * 8 + 7] = __uint_as_float(bx | (by << 8) | (bz << 16) | (b << 24));
    }
    __syncthreads();

    // --- scatter loop: one (atom, offset) pair per thread per iteration -----
    const int npairs = atoms * K_OFF;
    for (int p = tid; p < npairs; p += SCATTER_THREADS) {
        const int i = p / K_OFF;          // atom within block
        const int k = p - i * K_OFF;      // offset index 0..124

        const float cx = s_atom[i * 8 + 0];
        const float cy = s_atom[i * 8 + 1];
        const float cz = s_atom[i * 8 + 2];
        const float r2 = s_atom[i * 8 + 3];
        const float f0 = s_atom[i * 8 + 4];
        const float f1 = s_atom[i * 8 + 5];
        const float f2 = s_atom[i * 8 + 6];
        const unsigned pk = __float_as_uint(s_atom[i * 8 + 7]);

        const int bx = (int)(pk & 0xffu);
        const int by = (int)((pk >> 8) & 0xffu);
        const int bz = (int)((pk >> 16) & 0xffu);
        const int b  = (int)(pk >> 24);

        // meshgrid('ij').ravel(): x slowest, z fastest
        const int ox = k / 25 - 2;
        const int rm = k % 25;
        const int oy = rm / 5 - 2;
        const int oz = rm % 5 - 2;

        const int vx = bx + ox, vy = by + oy, vz = bz + oz;

        const float dx = __fsub_rn((float)vx, cx);
        const float dy = __fsub_rn((float)vy, cy);
        const float dz = __fsub_rn((float)vz, cz);
        const float d2 = __fadd_rn(__fadd_rn(__fmul_rn(dx, dx), __fmul_rn(dy, dy)),
                                   __fmul_rn(dz, dz));

        const bool inb = ((unsigned)vx < (unsigned)VOL) &
                         ((unsigned)vy < (unsigned)VOL) &
                         ((unsigned)vz < (unsigned)VOL);

        if (inb && (d2 <= r2)) {
            const int lin = b * (VOL * VOL * VOL) + ((vx * VOL + vy) * VOL + vz);
            float* o = out + (size_t)lin * 3;
            __hip_atomic_fetch_max(o + 0, f0, __ATOMIC_RELAXED, __HIP_MEMORY_SCOPE_AGENT);
            __hip_atomic_fetch_min(o + 1, f1, __ATOMIC_RELAXED, __HIP_MEMORY_SCOPE_AGENT);
            __hip_atomic_fetch_add(o + 2, f2, __ATOMIC_RELAXED, __HIP_MEMORY_SCOPE_AGENT);
            __hip_atomic_fetch_add(cnt + lin, 1u, __ATOMIC_RELAXED, __HIP_MEMORY_SCOPE_AGENT);
        }
    }
}

// ---------------------------------------------------------------- init ------
__global__ void vdw_init_kernel(float* __restrict__ out,
                                unsigned* __restrict__ cnt,
                                int total) {
    int i = blockIdx.x * blockDim.x + threadIdx.x;
    if (i >= total) return;
    out[3 * i + 0] = -__builtin_inff();
    out[3 * i + 1] =  __builtin_inff();
    out[3 * i + 2] = 0.0f;
    cnt[i] = 0u;
}

// ------------------------------------------------------------ finalize ------
__global__ void vdw_finalize_kernel(float* __restrict__ out,
                                    const unsigned* __restrict__ cnt,
                                    int total) {
    int i = blockIdx.x * blockDim.x + threadIdx.x;
    if (i >= total) return;
    const unsigned c = cnt[i];
    const bool occ = (c > 0u);
    float mx = out[3 * i + 0];
    float mn = out[3 * i + 1];
    float s  = out[3 * i + 2];
    out[3 * i + 0] = occ ? mx : 0.0f;
    out[3 * i + 1] = occ ? mn : 0.0f;
    out[3 * i + 2] = __fdiv_rn(s, (float)(occ ? c : 1u));
}

// ------------------------------------------------------------- launch -------
extern "C" void kernel_launch(void* const* d_in, const int* in_sizes, int n_in,
                              void* d_out, int out_size, void* d_ws, size_t ws_size,
                              hipStream_t stream) {
    const float* cr = (const float*)d_in[0];   // (B,N,4) float32
    const float* ft = (const float*)d_in[1];   // (B,N,3) float32
    float*    out   = (float*)d_out;           // (B,V,V,V,3) float32
    unsigned* cnt   = (unsigned*)d_ws;         // B*V^3 counters

    const int total  = out_size / 3;           // B*V^3
    const int BN     = in_sizes[0] / 4;        // B*N
    const int B      = total / (VOL * VOL * VOL);
    const int NperB  = BN / (B > 0 ? B : 1);

    const int T = 256;
    vdw_init_kernel<<<(total + T - 1) / T, T, 0, stream>>>(out, cnt, total);

    const int blocks = (BN + ATOMS_PER_BLOCK - 1) / ATOMS_PER_BLOCK;
    vdw_scatter_kernel<<<blocks, SCATTER_THREADS, 0, stream>>>(cr, ft, out, cnt, NperB, BN);

    vdw_finalize_kernel<<<(total + T - 1) / T, T, 0, stream>>>(out, cnt, total);
}